// ncc_loss_1142461300850
// MI455X (gfx1250) — compile-verified
//
#include <hip/hip_runtime.h>
#include <math.h>

typedef __attribute__((ext_vector_type(2))) float v2f;
typedef __attribute__((ext_vector_type(8))) float v8f;

#define IMG_H 512
#define IMG_W 512
#define NBATCH 16
#define WIN 15
#define PADW 7
#define BT 64                 // 64x64 output pixels per workgroup
#define REG 78                // BT + WIN - 1 (halo region)
#define RSTR 78               // region row stride (78 mod 64 = 14, conflict-free column reads)
#define WAVES 8
#define BLOCK_THREADS 256
#define GRID_XY ((IMG_W / BT) * (IMG_H / BT))   // 64
#define NBLOCKS (NBATCH * GRID_XY)              // 1024

// ---------------------------------------------------------------------------
// Stage 1: fused 15x15 box sums via f32 WMMA banded matmuls + per-pixel NCC
// statistics + deterministic per-block partial reduction.
//
// Box filter as matmul: out = R * W with W[k][c] = 1 iff c <= k <= c+14
// (K dim padded 30 -> 32; pad taps are exact zeros on both A and B sides).
// Pass 1:  H(32x16)  = R(32x32)  * W   (horizontal sums, two 16-row halves)
// Pass 2:  U(16x16)  = H^T(16x32)* W   (vertical sums; U is the transposed
//                                       result tile, fine since we only sum it)
// ---------------------------------------------------------------------------
__global__ __launch_bounds__(BLOCK_THREADS)
void ncc_stage1(const float* __restrict__ I, const float* __restrict__ J,
                float* __restrict__ partial) {
  __shared__ float shI[REG * RSTR];       // 24336 B
  __shared__ float shJ[REG * RSTR];       // 24336 B
  __shared__ float shH[WAVES][32 * 16];   // 16384 B (per-wave pass-1 result)
  __shared__ float red[WAVES][3];

  const int tid  = threadIdx.x;
  const int wave = tid >> 5;
  const int lane = tid & 31;
  const int half = lane >> 4;   // lane half selects K sub-pair in A/B layouts
  const int ml   = lane & 15;   // M (A rows) / N (B,C,D cols) lane index

  const int blk = blockIdx.x;
  const int b   = blk / GRID_XY;
  const int rb  = blk % GRID_XY;
  const int by  = rb / (IMG_W / BT);
  const int bx  = rb % (IMG_W / BT);
  const int R0  = by * BT;
  const int C0  = bx * BT;

  const float* Ib = I + (size_t)b * IMG_H * IMG_W;
  const float* Jb = J + (size_t)b * IMG_H * IMG_W;

  // Cooperative halo-region load with zero padding (coalesced along columns).
  for (int e = tid; e < REG * RSTR; e += BLOCK_THREADS) {
    const int r  = e / RSTR;
    const int c  = e % RSTR;
    const int gr = R0 - PADW + r;
    const int gc = C0 - PADW + c;
    const bool inb = (gr >= 0) && (gr < IMG_H) && (gc >= 0) && (gc < IMG_W);
    const size_t gi = (size_t)gr * IMG_W + gc;
    shI[e] = inb ? Ib[gi] : 0.0f;
    shJ[e] = inb ? Jb[gi] : 0.0f;
  }
  __syncthreads();

  // Constant banded-ones B operand, 8 K-chunks of 4x16.
  // Assumed f32 B(4x16) layout (analogous to C/D upper-half rule):
  //   vgpr0 -> K = 4*k0 + 2*half, vgpr1 -> K = 4*k0 + 2*half + 1, N = ml.
  v2f wk[8];
#pragma unroll
  for (int k0 = 0; k0 < 8; ++k0) {
    const int kx = 4 * k0 + 2 * half;
    wk[k0].x = (kx     >= ml && kx     <= ml + (WIN - 1)) ? 1.0f : 0.0f;
    wk[k0].y = (kx + 1 >= ml && kx + 1 <= ml + (WIN - 1)) ? 1.0f : 0.0f;
  }

  float* Hb = shH[wave];
  float lc = 0.0f, li = 0.0f, lj = 0.0f;
  const float inv_n = 1.0f / (float)(WIN * WIN);
  const float n_f   = (float)(WIN * WIN);

  // 16 tiles of 16x16 per block; each wave handles two.
  for (int t = wave; t < 16; t += WAVES) {
    const int roff = 16 * (t >> 2);
    const int coff = 16 * (t & 3);

    // Clamped quantity read: region coords (rr,kk) relative to this tile,
    // valid range [0,30); K-pad positions contribute exact zeros.
    auto rdq = [&](int q, int rr, int kk) -> float {
      if (rr >= (16 + WIN - 1) || kk >= (16 + WIN - 1)) return 0.0f;
      const int idx = (roff + rr) * RSTR + (coff + kk);
      const float iv = shI[idx];
      const float jv = shJ[idx];
      switch (q) {
        case 0:  return iv;
        case 1:  return jv;
        case 2:  return iv * iv;
        case 3:  return jv * jv;
        default: return iv * jv;
      }
    };

    v8f acc[5];
#pragma unroll
    for (int q = 0; q < 5; ++q) {
      // Pass 1: horizontal sums. A(16x4) layout: lanes 0-15 -> K={0,1},
      // lanes 16-31 -> K={2,3}, row = ml.
#pragma unroll
      for (int h = 0; h < 2; ++h) {
        const int rr = 16 * h + ml;
        v8f vc = {};
#pragma unroll
        for (int k0 = 0; k0 < 8; ++k0) {
          const int kk = 4 * k0 + 2 * half;
          v2f a;
          a.x = rdq(q, rr, kk);
          a.y = rdq(q, rr, kk + 1);
          vc = __builtin_amdgcn_wmma_f32_16x16x4_f32(
              false, a, false, wk[k0], (short)0, vc, false, false);
        }
        // Spill D (C/D layout: M = i + 8*half, N = ml) to H rows 16h+M.
#pragma unroll
        for (int i = 0; i < 8; ++i)
          Hb[(16 * h + i + 8 * half) * 16 + ml] = vc[i];
      }

      // Pass 2: vertical sums on H^T; transpose realized by the LDS read
      // pattern (A row = output column ml, A K = H row).
      v8f u = {};
#pragma unroll
      for (int k0 = 0; k0 < 8; ++k0) {
        const int kk = 4 * k0 + 2 * half;
        v2f a;
        a.x = Hb[kk * 16 + ml];
        a.y = Hb[(kk + 1) * 16 + ml];
        u = __builtin_amdgcn_wmma_f32_16x16x4_f32(
            false, a, false, wk[k0], (short)0, u, false, false);
      }
      acc[q] = u;
    }

    // Per-pixel NCC statistics (faithful to reference expressions).
#pragma unroll
    for (int i = 0; i < 8; ++i) {
      const float s1 = acc[0][i], s2 = acc[1][i];
      const float s11 = acc[2][i], s22 = acc[3][i], s12 = acc[4][i];
      const float uI = s1 * inv_n, uJ = s2 * inv_n;
      lc += s12 - uJ * s1 - uI * s2 + uI * uJ * n_f;
      li += s11 - 2.0f * uI * s1 + uI * uI * n_f;
      lj += s22 - 2.0f * uJ * s2 + uJ * uJ * n_f;
    }
  }

  // Deterministic reduction: wave32 shuffle tree -> per-wave slots -> block.
#pragma unroll
  for (int m = 16; m >= 1; m >>= 1) {
    lc += __shfl_xor(lc, m, 32);
    li += __shfl_xor(li, m, 32);
    lj += __shfl_xor(lj, m, 32);
  }
  if (lane == 0) { red[wave][0] = lc; red[wave][1] = li; red[wave][2] = lj; }
  __syncthreads();
  if (tid == 0) {
    float sc = 0.0f, si = 0.0f, sj = 0.0f;
    for (int w = 0; w < WAVES; ++w) { sc += red[w][0]; si += red[w][1]; sj += red[w][2]; }
    partial[blk * 3 + 0] = sc;
    partial[blk * 3 + 1] = si;
    partial[blk * 3 + 2] = sj;
  }
}

// ---------------------------------------------------------------------------
// Stage 2: fixed-order final reduction over 1024 partial triples + scalar loss.
// ---------------------------------------------------------------------------
__global__ __launch_bounds__(256)
void ncc_stage2(const float* __restrict__ partial, float* __restrict__ out) {
  __shared__ float s[256][3];
  const int tid = threadIdx.x;
  float sc = 0.0f, si = 0.0f, sj = 0.0f;
  for (int i = tid; i < NBLOCKS; i += 256) {
    sc += partial[i * 3 + 0];
    si += partial[i * 3 + 1];
    sj += partial[i * 3 + 2];
  }
  s[tid][0] = sc; s[tid][1] = si; s[tid][2] = sj;
  __syncthreads();
  for (int st = 128; st >= 1; st >>= 1) {
    if (tid < st) {
      s[tid][0] += s[tid + st][0];
      s[tid][1] += s[tid + st][1];
      s[tid][2] += s[tid + st][2];
    }
    __syncthreads();
  }
  if (tid == 0) out[0] = -(s[0][0] / sqrtf(s[0][1] * s[0][2]));
}

extern "C" void kernel_launch(void* const* d_in, const int* in_sizes, int n_in,
                              void* d_out, int out_size, void* d_ws, size_t ws_size,
                              hipStream_t stream) {
  const float* I = (const float*)d_in[0];
  const float* J = (const float*)d_in[1];
  float* out     = (float*)d_out;
  float* partial = (float*)d_ws;   // 1024 * 3 floats = 12 KB scratch
  (void)in_sizes; (void)n_in; (void)out_size; (void)ws_size;

  ncc_stage1<<<NBLOCKS, BLOCK_THREADS, 0, stream>>>(I, J, partial);
  ncc_stage2<<<1, 256, 0, stream>>>(partial, out);
}